// GeometricLayer_27797028339958
// MI455X (gfx1250) — compile-verified
//
#include <hip/hip_runtime.h>

// ---------------------------------------------------------------------------
// GeometricLayer on MI455X (gfx1250), fp32 path via V_WMMA_F32_16X16X4_F32.
//
// out[(r,c),:] = X[r,c]@W0 + X[c,r]@W1 + RV[r] + CV[c] + (r==c)*DV[r]
// ---------------------------------------------------------------------------

typedef __attribute__((ext_vector_type(2))) float v2f;
typedef __attribute__((ext_vector_type(8))) float v8f;
typedef __attribute__((__vector_size__(16))) int v4i_;   // matches builtin param type

#if defined(__has_builtin)
#if __has_builtin(__builtin_amdgcn_global_load_async_to_lds_b128)
#define USE_ASYNC_LDS 1
#endif
#endif

static constexpr int N   = 1024;
static constexpr int K   = 64;
static constexpr int OUT = 64;

__device__ __forceinline__ float4 f4add(float4 a, float4 b) {
  return make_float4(a.x + b.x, a.y + b.y, a.z + b.z, a.w + b.w);
}

// ---------------------------------------------------------------------------
// Stage 1: cp[r] = mean_c X[r,c], rp[c] = mean_r X[r,c]   (1024 blocks x 256)
// float4-vectorized: thread = (stripe of 64 grid entries) x (float4 k-group)
// ---------------------------------------------------------------------------
__global__ __launch_bounds__(256) void pool_kernel(const float* __restrict__ X,
                                                   float* __restrict__ rp,
                                                   float* __restrict__ cp) {
  __shared__ float4 red[256];
  const int b = blockIdx.x;
  const int t = threadIdx.x;
  const int kq = t & 15;           // float4 group within K (k = kq*4..kq*4+3)
  const int stripe = t >> 4;       // 16 stripes of 64 grid entries
  const float4* Xv = (const float4*)X;   // 16 float4 per grid-element feature row

  // ---- cp[b]: row band b (contiguous 256 KB) ----
  float4 acc = make_float4(0.f, 0.f, 0.f, 0.f);
  for (int c = stripe * 64; c < stripe * 64 + 64; ++c)
    acc = f4add(acc, Xv[((size_t)b * N + c) * 16 + kq]);
  red[t] = acc;
  __syncthreads();
  if (t < 128) red[t] = f4add(red[t], red[t + 128]);
  __syncthreads();
  if (t < 64)  red[t] = f4add(red[t], red[t + 64]);
  __syncthreads();
  if (t < 32)  red[t] = f4add(red[t], red[t + 32]);
  __syncthreads();
  if (t < 16) {
    float4 v = f4add(red[t], red[t + 16]);
    const float s = 1.0f / N;
    ((float4*)cp)[b * 16 + t] = make_float4(v.x * s, v.y * s, v.z * s, v.w * s);
  }
  __syncthreads();

  // ---- rp[b]: grid column b (1024 rows of 256 B, stride 256 KB) ----
  acc = make_float4(0.f, 0.f, 0.f, 0.f);
  for (int r = stripe * 64; r < stripe * 64 + 64; ++r)
    acc = f4add(acc, Xv[((size_t)r * N + b) * 16 + kq]);
  red[t] = acc;
  __syncthreads();
  if (t < 128) red[t] = f4add(red[t], red[t + 128]);
  __syncthreads();
  if (t < 64)  red[t] = f4add(red[t], red[t + 64]);
  __syncthreads();
  if (t < 32)  red[t] = f4add(red[t], red[t + 32]);
  __syncthreads();
  if (t < 16) {
    float4 v = f4add(red[t], red[t + 16]);
    const float s = 1.0f / N;
    ((float4*)rp)[b * 16 + t] = make_float4(v.x * s, v.y * s, v.z * s, v.w * s);
  }
}

// ---------------------------------------------------------------------------
// Stage 2: block 0: ap = mean_r cp[r];  block 1: dp = mean_i X[i,i]
// ---------------------------------------------------------------------------
__global__ __launch_bounds__(256) void reduce_kernel(const float* __restrict__ X,
                                                     const float* __restrict__ cp,
                                                     float* __restrict__ ap,
                                                     float* __restrict__ dp) {
  __shared__ float4 red[256];
  const int t = threadIdx.x;
  const int kq = t & 15;
  const int stripe = t >> 4;
  float4 acc = make_float4(0.f, 0.f, 0.f, 0.f);
  if (blockIdx.x == 0) {
    const float4* cpv = (const float4*)cp;
    for (int i = stripe * 64; i < stripe * 64 + 64; ++i)
      acc = f4add(acc, cpv[i * 16 + kq]);
  } else {
    const float4* Xv = (const float4*)X;
    for (int i = stripe * 64; i < stripe * 64 + 64; ++i)
      acc = f4add(acc, Xv[((size_t)i * (N + 1)) * 16 + kq]);
  }
  red[t] = acc;
  __syncthreads();
  if (t < 128) red[t] = f4add(red[t], red[t + 128]);
  __syncthreads();
  if (t < 64)  red[t] = f4add(red[t], red[t + 64]);
  __syncthreads();
  if (t < 32)  red[t] = f4add(red[t], red[t + 32]);
  __syncthreads();
  if (t < 16) {
    float4 v = f4add(red[t], red[t + 16]);
    const float s = 1.0f / N;
    float4* dst = (float4*)(blockIdx.x == 0 ? ap : dp);
    dst[t] = make_float4(v.x * s, v.y * s, v.z * s, v.w * s);
  }
}

// ---------------------------------------------------------------------------
// Stage 3: RV / CV / DV tables, small GEMMs (~60 MFLOP total, scalar is fine)
// weights layout: W[(w*K + k)*OUT + o]
// ---------------------------------------------------------------------------
__global__ __launch_bounds__(256) void combine_kernel(const float* __restrict__ X,
                                                      const float* __restrict__ W,
                                                      const float* __restrict__ rp,
                                                      const float* __restrict__ cp,
                                                      const float* __restrict__ ap,
                                                      const float* __restrict__ dp,
                                                      float* __restrict__ RV,
                                                      float* __restrict__ CV,
                                                      float* __restrict__ DV) {
  const int tid = blockIdx.x * 256 + threadIdx.x;  // N*OUT threads
  const int i = tid >> 6;
  const int o = tid & 63;
  const float* diag_i = X + (size_t)i * (N + 1) * K;
  float rv = 0.0f, cv = 0.0f, dv = 0.0f;
  for (int kx = 0; kx < K; ++kx) {
    const float rpv = rp[i * K + kx];
    const float cpv = cp[i * K + kx];
    const float dg  = diag_i[kx];
    const float apv = ap[kx];
    const float dpv = dp[kx];
    rv += rpv * W[( 4 * K + kx) * OUT + o];   // (rp@W4)[r]
    rv += cpv * W[( 5 * K + kx) * OUT + o];   // (cp@W5)[r]
    rv += dg  * W[(14 * K + kx) * OUT + o];   // (diag@W14)[r]
    rv += apv * W[( 7 * K + kx) * OUT + o];   // ap@W7 (constant, folded)
    rv += dpv * W[(12 * K + kx) * OUT + o];   // dp@W12 (constant, folded)
    cv += rpv * W[( 3 * K + kx) * OUT + o];
    cv += cpv * W[( 6 * K + kx) * OUT + o];
    cv += dg  * W[(13 * K + kx) * OUT + o];
    dv += dg  * W[( 2 * K + kx) * OUT + o];
    dv += rpv * W[(10 * K + kx) * OUT + o];
    dv += cpv * W[(11 * K + kx) * OUT + o];
    dv += dpv * W[( 8 * K + kx) * OUT + o];
    dv += apv * W[( 9 * K + kx) * OUT + o];
  }
  RV[tid] = rv;
  CV[tid] = cv;
  DV[tid] = dv;
}

// ---------------------------------------------------------------------------
// Stage 4: main WMMA pass. 2080 blocks = triangular pairs (rb<=cb); block
// handles output tiles (rb,cb) and (cb,rb) so each X tile is read once.
// LDS: two 16x16x64 fp32 X tiles (64 KB each) + W0,W1 (32 KB) = 160 KB.
// Staging uses GLOBAL_LOAD_ASYNC_TO_LDS_B128 when available.
// ---------------------------------------------------------------------------
__global__ __launch_bounds__(256) void main_kernel(const float* __restrict__ X,
                                                   const float* __restrict__ W,
                                                   const float* __restrict__ RV,
                                                   const float* __restrict__ CV,
                                                   const float* __restrict__ DV,
                                                   float* __restrict__ out) {
  __shared__ float ldsA[256 * 64];    // grid tile (rb, cb): row (j*16+i) = X[rb*16+j, cb*16+i]
  __shared__ float ldsB[256 * 64];    // grid tile (cb, rb)
  __shared__ float ldsW[2 * 64 * 64]; // W0 then W1

  // triangular decode: block p -> (rb, cb), rb <= cb, T = 64
  const int p = blockIdx.x;
  int rb = (int)((129.0f - sqrtf(16641.0f - 8.0f * (float)p)) * 0.5f);
  while (rb * (129 - rb) / 2 > p) --rb;                 // exact correction
  while ((rb + 1) * (129 - (rb + 1)) / 2 <= p) ++rb;
  const int cb = rb + (p - rb * (129 - rb) / 2);

  const int t = threadIdx.x;
  const int jj = t >> 4, ii = t & 15;
  const size_t rowA = ((size_t)(rb * 16 + jj) * N + cb * 16 + ii) * K;
  const size_t rowB = ((size_t)(cb * 16 + jj) * N + rb * 16 + ii) * K;

#ifdef USE_ASYNC_LDS
  {
    v4i_* srcA = (v4i_*)(X + rowA);       // 16B chunks of this thread's 256B row
    v4i_* srcB = (v4i_*)(X + rowB);
    v4i_* dA   = (v4i_*)(ldsA + t * 64);
    v4i_* dB   = (v4i_*)(ldsB + t * 64);
#pragma unroll
    for (int q = 0; q < 16; ++q) {
      __builtin_amdgcn_global_load_async_to_lds_b128(srcA + q, dA + q, 0, 0);
      __builtin_amdgcn_global_load_async_to_lds_b128(srcB + q, dB + q, 0, 0);
    }
    // weights: 512 x 16B chunks across 256 threads
    v4i_* srcW = (v4i_*)W;
    v4i_* dW   = (v4i_*)ldsW;
#pragma unroll
    for (int q = 0; q < 2; ++q)
      __builtin_amdgcn_global_load_async_to_lds_b128(srcW + t + q * 256,
                                                     dW + t + q * 256, 0, 0);
    asm volatile("s_wait_asynccnt 0" ::: "memory");
  }
#else
  {
    for (int q = t; q < 2 * 64 * 64; q += 256) ldsW[q] = W[q];
    const float4* srcA = (const float4*)(X + rowA);
    const float4* srcB = (const float4*)(X + rowB);
    float4* dA = (float4*)(ldsA + t * 64);
    float4* dB = (float4*)(ldsB + t * 64);
#pragma unroll
    for (int q = 0; q < 16; ++q) { dA[q] = srcA[q]; dB[q] = srcB[q]; }
  }
#endif
  __syncthreads();

  const int wave = t >> 5;
  const int lane = t & 31;
  const int m16  = lane & 15;        // M (or N) index within fragment
  const int half = lane >> 4;        // K-half selector per ISA 16x4 f32 layout
  const int nst  = (rb == cb) ? 16 : 32;   // s-tiles: 16 per output grid tile

  for (int st = wave; st < nst; st += 8) {
    const int tsel = st >> 4;        // 0 -> out tile (rb,cb), 1 -> out tile (cb,rb)
    const int j = st & 15;           // local r within output tile
    const float* A1 = tsel ? ldsB : ldsA;   // rows X[r, C*16 + m]
    const float* A2 = tsel ? ldsA : ldsB;   // rows X[C*16 + m, r]  (transpose term)
    const int R = tsel ? cb : rb;
    const int C = tsel ? rb : cb;

    v8f acc[4] = {v8f{}, v8f{}, v8f{}, v8f{}};

#pragma unroll
    for (int kk = 0; kk < 16; ++kk) {
      const int kbase = kk * 4 + half * 2; // A/B fragment: VGPR v holds K = kbase+v
      v2f a1, a2;
      a1.x = A1[(j * 16 + m16) * 64 + kbase];
      a1.y = A1[(j * 16 + m16) * 64 + kbase + 1];
      a2.x = A2[(m16 * 16 + j) * 64 + kbase];
      a2.y = A2[(m16 * 16 + j) * 64 + kbase + 1];
#pragma unroll
      for (int nc = 0; nc < 4; ++nc) {
        v2f b0, b1;
        b0.x = ldsW[(kbase    ) * 64 + nc * 16 + m16];
        b0.y = ldsW[(kbase + 1) * 64 + nc * 16 + m16];
        b1.x = ldsW[4096 + (kbase    ) * 64 + nc * 16 + m16];
        b1.y = ldsW[4096 + (kbase + 1) * 64 + nc * 16 + m16];
        acc[nc] = __builtin_amdgcn_wmma_f32_16x16x4_f32(
            false, a1, false, b0, (short)0, acc[nc], false, false);
        acc[nc] = __builtin_amdgcn_wmma_f32_16x16x4_f32(
            false, a2, false, b1, (short)0, acc[nc], false, false);
      }
    }

    // epilogue: D VGPR v -> (M = v + 8*half, N = m16); M is local c, N is out feature
    const int r = R * 16 + j;
#pragma unroll
    for (int nc = 0; nc < 4; ++nc) {
      const int o = nc * 16 + m16;
      const float rvv = RV[r * OUT + o];
      const float dvv = DV[r * OUT + o];   // uniform address; used only when c==r
#pragma unroll
      for (int v = 0; v < 8; ++v) {
        const int c = C * 16 + v + 8 * half;
        float val = acc[nc][v] + rvv + CV[c * OUT + o];
        val += (c == r) ? dvv : 0.0f;
        out[((size_t)r * N + c) * OUT + o] = val;
      }
    }
  }
}

// ---------------------------------------------------------------------------
// Launcher. Workspace layout (floats): rp[N*K] cp[N*K] ap[K] dp[K]
//                                      RV[N*OUT] CV[N*OUT] DV[N*OUT]  (~1.3 MB)
// ---------------------------------------------------------------------------
extern "C" void kernel_launch(void* const* d_in, const int* in_sizes, int n_in,
                              void* d_out, int out_size, void* d_ws, size_t ws_size,
                              hipStream_t stream) {
  (void)in_sizes; (void)n_in; (void)out_size; (void)ws_size;
  const float* X = (const float*)d_in[0];   // input_layer [S, 64]
  const float* W = (const float*)d_in[1];   // weights [15, 64, 64]
  float* out = (float*)d_out;

  float* rp = (float*)d_ws;
  float* cp = rp + N * K;
  float* ap = cp + N * K;
  float* dp = ap + K;
  float* RV = dp + K;
  float* CV = RV + N * OUT;
  float* DV = CV + N * OUT;

  pool_kernel<<<N, 256, 0, stream>>>(X, rp, cp);
  reduce_kernel<<<2, 256, 0, stream>>>(X, cp, ap, dp);
  combine_kernel<<<(N * OUT) / 256, 256, 0, stream>>>(X, W, rp, cp, ap, dp, RV, CV, DV);
  const int npairs = 64 * 65 / 2;   // 2080 triangular tile pairs
  main_kernel<<<npairs, 256, 0, stream>>>(X, W, RV, CV, DV, out);
}